// LSTM_8589935226
// MI455X (gfx1250) — compile-verified
//
#include <hip/hip_runtime.h>

// ---------------------------------------------------------------------------
// Fused tiny-LSTM (B=4.19M, T=4, I=1, H=2, C=1) for MI455X / gfx1250.
//
// Roofline: 84 MB traffic -> ~3.6us at 23.3 TB/s; but ~38 tanh-class TRANS
// ops per element dominate (~10us class). gfx1250 hardware v_tanh_f32 keeps
// every sigmoid/tanh at exactly 1 TRANS op. Weights live in SGPRs
// (s_load_b512 nv), x streams as non-temporal b128, outputs as NT b32.
// WMMA deliberately NOT used: the per-step matmul is (2x8) with a serial
// T-dependence; a 16x16x4 tile is >75% padding and doubles TRANS work on
// garbage lanes. This is the correct instruction mix for this shape.
// ---------------------------------------------------------------------------

typedef float v4f __attribute__((ext_vector_type(4)));

#define NGATES  8   // 4*H, H=2, PyTorch gate order: i, f, g, o
#define T_STEPS 4
#define ELEMS   8   // batch elements per thread

__device__ __forceinline__ float tanh_fast(float x) {
#if __has_builtin(__builtin_amdgcn_tanhf)
  return __builtin_amdgcn_tanhf(x);            // v_tanh_f32 (gfx1250 trans op)
#elif __has_builtin(__builtin_amdgcn_tanh_f32)
  return __builtin_amdgcn_tanh_f32(x);
#else
  float e = __expf(2.0f * x);                  // v_exp_f32 fallback
  return 1.0f - 2.0f / (e + 1.0f);
#endif
}

__device__ __forceinline__ float sigmoid_fast(float x) {
  // sigmoid(x) = 0.5 + 0.5*tanh(x/2): exactly one TRANS op
  return __builtin_fmaf(0.5f, tanh_fast(0.5f * x), 0.5f);
}

// One batch element: 4-step LSTM + FC head. xv = {x_t0..x_t3}.
__device__ __forceinline__ float lstm_elem(
    v4f xv, const float* __restrict__ wih, const float* __restrict__ whh,
    const float* __restrict__ bias, float wfc0, float wfc1, float bfc) {
  const float xs[4] = {xv.x, xv.y, xv.z, xv.w};
  float h0, h1, c0, c1;

  // ---- t = 0 specialization: h=c=0 => forget gate is dead code ----
  {
    float gi0 = __builtin_fmaf(xs[0], wih[0], bias[0]);
    float gi1 = __builtin_fmaf(xs[0], wih[1], bias[1]);
    float gg0 = __builtin_fmaf(xs[0], wih[4], bias[4]);
    float gg1 = __builtin_fmaf(xs[0], wih[5], bias[5]);
    float go0 = __builtin_fmaf(xs[0], wih[6], bias[6]);
    float go1 = __builtin_fmaf(xs[0], wih[7], bias[7]);
    float i0 = sigmoid_fast(gi0), i1 = sigmoid_fast(gi1);
    float g0 = tanh_fast(gg0),    g1 = tanh_fast(gg1);
    float o0 = sigmoid_fast(go0), o1 = sigmoid_fast(go1);
    c0 = i0 * g0;
    c1 = i1 * g1;
    h0 = o0 * tanh_fast(c0);
    h1 = o1 * tanh_fast(c1);
  }

  // ---- t = 1..3 ----
#pragma unroll
  for (int t = 1; t < T_STEPS; ++t) {
    float g[NGATES];
#pragma unroll
    for (int j = 0; j < NGATES; ++j) {
      float v = __builtin_fmaf(xs[t], wih[j], bias[j]);      // x-projection
      v = __builtin_fmaf(h0, whh[2 * j + 0], v);             // h @ W_hh^T
      g[j] = __builtin_fmaf(h1, whh[2 * j + 1], v);
    }
    float i0 = sigmoid_fast(g[0]), i1 = sigmoid_fast(g[1]);
    float f0 = sigmoid_fast(g[2]), f1 = sigmoid_fast(g[3]);
    float t0 = tanh_fast(g[4]),    t1 = tanh_fast(g[5]);
    float o0 = sigmoid_fast(g[6]), o1 = sigmoid_fast(g[7]);
    c0 = __builtin_fmaf(f0, c0, i0 * t0);
    c1 = __builtin_fmaf(f1, c1, i1 * t1);
    h0 = o0 * tanh_fast(c0);
    h1 = o1 * tanh_fast(c1);
  }

  // ---- FC head: out = h @ W_fc^T + b_fc ----
  return __builtin_fmaf(h1, wfc1, __builtin_fmaf(h0, wfc0, bfc));
}

__global__ __launch_bounds__(256) void lstm_fused_kernel(
    const float* __restrict__ x,     // [B, T, I] = [B,4,1]: one float4 per b
    const float* __restrict__ W_ih,  // [8, 1]
    const float* __restrict__ W_hh,  // [8, 2] row-major: whh[2*j+k]
    const float* __restrict__ b_ih,  // [8]
    const float* __restrict__ b_hh,  // [8]
    const float* __restrict__ W_fc,  // [1, 2]
    const float* __restrict__ b_fc,  // [1]
    float* __restrict__ out,         // [B, 1]
    int B) {
  // ---- wave-uniform weights: 43 scalars -> SGPRs (s_load_b512 nv) ----
  float wih[NGATES], whh[2 * NGATES], bias[NGATES];
  {
    const v4f* wih4 = (const v4f*)W_ih;
    const v4f* whh4 = (const v4f*)W_hh;
    const v4f* bih4 = (const v4f*)b_ih;
    const v4f* bhh4 = (const v4f*)b_hh;
    v4f a0 = wih4[0], a1 = wih4[1];
    v4f w0 = whh4[0], w1 = whh4[1], w2 = whh4[2], w3 = whh4[3];
    v4f p0 = bih4[0], p1 = bih4[1];
    v4f q0 = bhh4[0], q1 = bhh4[1];
#pragma unroll
    for (int k = 0; k < 4; ++k) {
      wih[k]      = a0[k];
      wih[4 + k]  = a1[k];
      whh[k]      = w0[k];
      whh[4 + k]  = w1[k];
      whh[8 + k]  = w2[k];
      whh[12 + k] = w3[k];
      bias[k]     = p0[k] + q0[k];   // b_ih + b_hh folded once
      bias[4 + k] = p1[k] + q1[k];
    }
  }
  const float wfc0 = W_fc[0], wfc1 = W_fc[1], bfc = b_fc[0];

  const int tid  = blockIdx.x * blockDim.x + threadIdx.x;
  const int nthr = gridDim.x * blockDim.x;
  const v4f* x4  = (const v4f*)x;

  if (tid + (ELEMS - 1) * nthr < B) {
    // ---- fast path: all ELEMS elements in range, no per-element checks ----
#pragma unroll
    for (int e = 0; e < ELEMS; ++e) {
      const int b = tid + e * nthr;  // lane-consecutive -> coalesced b128/b32
      if (e + 1 < ELEMS)             // global_prefetch_b8 for next element
        __builtin_prefetch((const void*)&x4[b + nthr], 0, 0);
      v4f xv = __builtin_nontemporal_load(&x4[b]);
      float r = lstm_elem(xv, wih, whh, bias, wfc0, wfc1, bfc);
      __builtin_nontemporal_store(r, &out[b]);
    }
  } else {
    // ---- ragged tail: guarded ----
#pragma unroll
    for (int e = 0; e < ELEMS; ++e) {
      const int b = tid + e * nthr;
      if (b >= B) continue;
      v4f xv = __builtin_nontemporal_load(&x4[b]);
      float r = lstm_elem(xv, wih, whh, bias, wfc0, wfc1, bfc);
      __builtin_nontemporal_store(r, &out[b]);
    }
  }
}

extern "C" void kernel_launch(void* const* d_in, const int* in_sizes, int n_in,
                              void* d_out, int out_size, void* d_ws, size_t ws_size,
                              hipStream_t stream) {
  (void)n_in; (void)out_size; (void)d_ws; (void)ws_size;
  const float* x    = (const float*)d_in[0];
  const float* W_ih = (const float*)d_in[1];
  const float* W_hh = (const float*)d_in[2];
  const float* b_ih = (const float*)d_in[3];
  const float* b_hh = (const float*)d_in[4];
  const float* W_fc = (const float*)d_in[5];
  const float* b_fc = (const float*)d_in[6];
  float* out = (float*)d_out;

  const int B = in_sizes[0] / (T_STEPS * 1);   // x is [B, T, I], T*I = 4

  const int threads_total = (B + ELEMS - 1) / ELEMS;
  const int block = 256;                        // 8 wave32 per workgroup
  const int grid  = (threads_total + block - 1) / block;   // 2048 for B=4.19M

  hipLaunchKernelGGL(lstm_fused_kernel, dim3(grid), dim3(block), 0, stream,
                     x, W_ih, W_hh, b_ih, b_hh, W_fc, b_fc, out, B);
}